// MessagePassing_9887014715655
// MI455X (gfx1250) — compile-verified
//
#include <hip/hip_runtime.h>
#include <stdint.h>

// ---------------------------------------------------------------------------
// MessagePassing: aggr[n] = x[n] * sum_{e: target[e]==n} W[e]
//   Gather and scatter index are identical -> the feature multiply distributes
//   out of the segment sum: 32x fewer atomics than the naive scatter.
//
// Pass 0: zero wsum[N]                                  (d_ws scratch)
// Pass A: wsum[target[e]] += W[e]
//         edge tiles streamed Global->LDS with GLOBAL_LOAD_ASYNC_TO_LDS_B128,
//         double-buffered, per-wave s_wait_asynccnt pipelining (CDNA5 path)
// Pass B: out[n*32+d] = x[n*32+d] * wsum[n]             (float4 streaming)
// ---------------------------------------------------------------------------

#define MP_BLOCK 256
#define MP_TILE  1024   // edges per tile: W 4KB + targets 8KB per buffer

__global__ void mp_zero_kernel(float* __restrict__ wsum, int n) {
    int i = blockIdx.x * blockDim.x + threadIdx.x;
    if (i < n) wsum[i] = 0.0f;
}

__global__ __launch_bounds__(MP_BLOCK)
void mp_edge_accum_async(const long long* __restrict__ target,
                         const float* __restrict__ W,
                         float* __restrict__ wsum,
                         int E, int fullTiles) {
    // Double-buffered staging: 2 * (8KB targets + 4KB weights) = 24KB LDS.
    __shared__ long long lds_t[2][MP_TILE];
    __shared__ float     lds_w[2][MP_TILE];

    const int tid    = threadIdx.x;
    const int stride = gridDim.x;

    // LDS byte offsets for this thread's 4-edge slot in each buffer.
    // Generic shared pointers carry the LDS offset in their low 32 bits
    // (ISA: LDS aperture keyed on ADDR[63:32]; LDS address = low 32 bits),
    // matching the async op's per-lane VGPR LDS-address operand.
    uint32_t lw[2], lt[2];
    lw[0] = (uint32_t)(uintptr_t)&lds_w[0][tid * 4];
    lw[1] = (uint32_t)(uintptr_t)&lds_w[1][tid * 4];
    lt[0] = (uint32_t)(uintptr_t)&lds_t[0][tid * 4];
    lt[1] = (uint32_t)(uintptr_t)&lds_t[1][tid * 4];

    // Stage one tile into buffer `buf`: 3 async b128 copies per thread
    // (16B of W, 2x16B of target). INST_OFFSET applies to BOTH the LDS and
    // the global address, so offset:16 fetches the second target chunk with
    // the same VGPR operands. Tracked by ASYNCcnt, in-order per wave.
    auto stage = [&](int tile, int buf) {
        uint32_t goff_w = (uint32_t)(tile * MP_TILE + tid * 4) * 4u;
        uint32_t goff_t = (uint32_t)(tile * MP_TILE + tid * 4) * 8u;
        asm volatile(
            "global_load_async_to_lds_b128 %0, %1, %4\n\t"
            "global_load_async_to_lds_b128 %2, %3, %5\n\t"
            "global_load_async_to_lds_b128 %2, %3, %5 offset:16"
            :
            : "v"(lw[buf]), "v"(goff_w),
              "v"(lt[buf]), "v"(goff_t),
              "s"(W), "s"(target)
            : "memory");
    };

    // Drain this thread's 4 staged edges: LDS reads + scalar f32 atomics.
    auto process = [&](int buf) {
        const longlong2* tp =
            reinterpret_cast<const longlong2*>(&lds_t[buf][tid * 4]);
        const longlong2 tA = tp[0];
        const longlong2 tB = tp[1];
        const float4 w =
            *reinterpret_cast<const float4*>(&lds_w[buf][tid * 4]);
        atomicAdd(&wsum[(int)tA.x], w.x);
        atomicAdd(&wsum[(int)tA.y], w.y);
        atomicAdd(&wsum[(int)tB.x], w.z);
        atomicAdd(&wsum[(int)tB.y], w.w);
    };

    int t = blockIdx.x;
    if (t < fullTiles) {
        int buf = 0;
        stage(t, 0);
        for (; t < fullTiles; t += stride) {
            const int  nt   = t + stride;
            const bool more = nt < fullTiles;
            if (more) {
                stage(nt, buf ^ 1);  // keep next tile in flight
                // 3 newer ops outstanding -> <=3 means current tile landed.
                asm volatile("s_wait_asynccnt 0x3" ::: "memory");
            } else {
                asm volatile("s_wait_asynccnt 0x0" ::: "memory");
            }
            process(buf);
            buf ^= 1;
        }
    }

    // Tail edges (E % MP_TILE): scalar path, block 0 only.
    if (blockIdx.x == 0) {
        for (int r = fullTiles * MP_TILE + tid; r < E; r += MP_BLOCK) {
            atomicAdd(&wsum[(int)target[r]], W[r]);
        }
    }
}

__global__ void mp_scale_kernel(const float* __restrict__ x,
                                const float* __restrict__ wsum,
                                float* __restrict__ out, int total4) {
    // total4 = N * 8  (8 float4 per 32-float feature row)
    int i = blockIdx.x * blockDim.x + threadIdx.x;
    if (i < total4) {
        const float4 v = reinterpret_cast<const float4*>(x)[i];
        const float  s = wsum[i >> 3];  // row scalar; hot in L0/L2
        float4 r;
        r.x = v.x * s;
        r.y = v.y * s;
        r.z = v.z * s;
        r.w = v.w * s;
        reinterpret_cast<float4*>(out)[i] = r;
    }
}

extern "C" void kernel_launch(void* const* d_in, const int* in_sizes, int n_in,
                              void* d_out, int out_size, void* d_ws, size_t ws_size,
                              hipStream_t stream) {
    // Inputs (setup_inputs order): edge_index (2,E) int64, x (N,32) f32, W (E,) f32
    const long long* edge_index = (const long long*)d_in[0];
    const float*     x          = (const float*)d_in[1];
    const float*     W          = (const float*)d_in[2];
    float*           out        = (float*)d_out;
    float*           wsum       = (float*)d_ws;   // N floats of scratch

    const int E = in_sizes[2];
    const int N = in_sizes[1] / 32;

    // Row 1 of edge_index is the target (both gather and scatter index).
    const long long* target = edge_index + (size_t)E;

    // Pass 0: zero the per-node weight accumulator.
    mp_zero_kernel<<<(N + MP_BLOCK - 1) / MP_BLOCK, MP_BLOCK, 0, stream>>>(wsum, N);

    // Pass A: async-staged scalar scatter-add of edge weights.
    {
        const int fullTiles = E / MP_TILE;
        int blocks = fullTiles;
        if (blocks > 512) blocks = 512;  // >=2 tiles/block -> double buffer pays off
        if (blocks < 1)   blocks = 1;    // still runs the scalar tail
        mp_edge_accum_async<<<blocks, MP_BLOCK, 0, stream>>>(target, W, wsum,
                                                             E, fullTiles);
    }

    // Pass B: out[n,:] = x[n,:] * wsum[n], float4-vectorized streaming pass.
    {
        const int total4 = N * 8;  // 32 floats per row = 8 float4
        mp_scale_kernel<<<(total4 + MP_BLOCK - 1) / MP_BLOCK, MP_BLOCK, 0, stream>>>(
            x, wsum, out, total4);
    }
}